// JacobiConv_68762426409166
// MI455X (gfx1250) — compile-verified
//
#include <hip/hip_runtime.h>
#include <cstdint>
#include <cstddef>

// ---------------------------------------------------------------------------
// JacobiConv for MI455X (gfx1250): 3x SpMM (edge-parallel, f32 global atomics,
// async-LDS-staged edge stream) + fused Jacobi recurrence / weighted mean.
// Memory-bound (AI ~0.3 flop/B); feature matrices (25.6MB) are L2-resident.
// CDNA5 paths used: global_load_async_to_lds_b128, s_wait_asynccnt,
// global_prefetch_b8 (__builtin_prefetch), global_atomic_add_f32 (wave32).
// ---------------------------------------------------------------------------

#define D_FEAT   64
#define TILE_E   1024
#define BLOCK    256
#define NWAVE    (BLOCK / 32)

// Jacobi recurrence constants (ALPHA = BETA = 1.0 from the reference)
constexpr double JA = 1.0, JB = 1.0;
constexpr double jc0(int l) { return 2.0 * l * (l + JA + JB) * (2 * l + JA + JB - 2); }
constexpr double jc1(int l) { return (2 * l + JA + JB - 1) * (JA * JA - JB * JB); }
constexpr double jc2(int l) { return (2 * l + JA + JB - 1) * (2 * l + JA + JB) * (2 * l + JA + JB - 2); }
constexpr double jc3(int l) { return 2.0 * (l + JA - 1) * (l + JB - 1) * (2 * l + JA + JB); }
constexpr float SCALING_F = 3.0f;

// --- CDNA5 async copy helpers (inline asm; portable across both toolchains) ---
__device__ __forceinline__ void async_b128_to_lds(void* lds_ptr, const void* gptr) {
    // Low 32 bits of the flat LDS pointer are the LDS byte offset (aperture
    // high bits are discarded by hardware for DS-space addresses).
    uint32_t lds_off = (uint32_t)(uintptr_t)lds_ptr;
    uint64_t gaddr   = (uint64_t)(uintptr_t)gptr;
    asm volatile("global_load_async_to_lds_b128 %0, %1, off"
                 :: "v"(lds_off), "v"(gaddr)
                 : "memory");
}
__device__ __forceinline__ void wait_async_zero() {
    asm volatile("s_wait_asynccnt 0" ::: "memory");
}

// ---------------------------------------------------------------------------
// SpMM: acc[row_e, :] += w_e * z[col_e, :]   (edge-parallel, 1 wave per edge)
// One block handles a TILE_E-edge tile; edge metadata is staged into LDS with
// async b128 loads, then each of the 8 waves walks edges with 32 lanes x
// float2 covering the 64-float feature row (256B fully coalesced).
// ---------------------------------------------------------------------------
__global__ __launch_bounds__(BLOCK) void spmm_kernel(
    const int* __restrict__ row, const int* __restrict__ col,
    const float* __restrict__ w, const float* __restrict__ z,
    float* __restrict__ acc, int nedges)
{
    __shared__ int   s_row[TILE_E];
    __shared__ int   s_col[TILE_E];
    __shared__ float s_w[TILE_E];

    const int tid  = threadIdx.x;
    const int lane = tid & 31;
    const int wv   = tid >> 5;

    const long long base = (long long)blockIdx.x * TILE_E;
    if (base >= nedges) return;
    int nrem = nedges - (int)base;
    if (nrem > TILE_E) nrem = TILE_E;

    // Stage this tile's edge stream into LDS (16B per thread per array).
    {
        const int i = tid * 4;
        if (i < nrem) {
            async_b128_to_lds(&s_row[i], row + base + i);
            async_b128_to_lds(&s_col[i], col + base + i);
            async_b128_to_lds(&s_w[i],   w   + base + i);
        }
        wait_async_zero();   // this wave's async transfers complete
    }
    __syncthreads();         // all waves' staged data visible

    for (int e = wv; e < nrem; e += NWAVE) {
        const int   c  = s_col[e];
        const int   r  = s_row[e];
        const float wt = s_w[e];

        const float2* zp = (const float2*)(z + (size_t)c * D_FEAT);
        float2 v = zp[lane];

        // Prefetch the next edge's feature row while we scatter this one.
        const int e2 = e + NWAVE;
        if (e2 < nrem) {
            __builtin_prefetch(z + (size_t)s_col[e2] * D_FEAT + lane * 2, 0, 3);
        }

        float* op = acc + (size_t)r * D_FEAT + lane * 2;
        unsafeAtomicAdd(op,     v.x * wt);   // global_atomic_add_f32 (no-return)
        unsafeAtomicAdd(op + 1, v.y * wt);
    }
}

// ---------------------------------------------------------------------------
// O = sa*A + sb*B + sc*C   (float4 vectorized)
// ---------------------------------------------------------------------------
__global__ __launch_bounds__(256) void combine3_kernel(
    const float4* __restrict__ A, const float4* __restrict__ B,
    const float4* __restrict__ C, float4* __restrict__ O,
    float sa, float sb, float sc, int n4)
{
    int i = blockIdx.x * blockDim.x + threadIdx.x;
    if (i >= n4) return;
    float4 a = A[i], b = B[i], c = C[i];
    float4 o;
    o.x = sa * a.x + sb * b.x + sc * c.x;
    o.y = sa * a.y + sb * b.y + sc * c.y;
    o.z = sa * a.z + sb * b.z + sc * c.z;
    o.w = sa * a.w + sb * b.w + sc * c.w;
    O[i] = o;
}

// ---------------------------------------------------------------------------
// Fused l=3 recurrence + cumprod(tanh(gamma)*SCALING)-weighted mean over l.
// out = ( coef0*x + coef1*z1 + coef2*z2 + coef3*z3 ) / 4,
// z3 = (jc1(3)*z2 + jc2(3)*Az - jc3(3)*z1) / jc0(3)
// ---------------------------------------------------------------------------
__global__ __launch_bounds__(256) void final_kernel(
    const float4* __restrict__ x,  const float4* __restrict__ z1,
    const float4* __restrict__ z2, const float4* __restrict__ Az,
    const float* __restrict__ gam, float4* __restrict__ out, int n4)
{
    int i = blockIdx.x * blockDim.x + threadIdx.x;
    if (i >= n4) return;

    const float coef0 = tanhf(gam[0]) * SCALING_F;
    const float coef1 = coef0 * tanhf(gam[1]) * SCALING_F;
    const float coef2 = coef1 * tanhf(gam[2]) * SCALING_F;
    const float coef3 = coef2 * tanhf(gam[3]) * SCALING_F;

    constexpr float f1 = (float)(jc1(3) / jc0(3));
    constexpr float f2 = (float)(jc2(3) / jc0(3));
    constexpr float f3 = (float)(jc3(3) / jc0(3));

    float4 X = x[i], Z1 = z1[i], Z2 = z2[i], AZ = Az[i];

    float4 Z3;
    Z3.x = f1 * Z2.x + f2 * AZ.x - f3 * Z1.x;
    Z3.y = f1 * Z2.y + f2 * AZ.y - f3 * Z1.y;
    Z3.z = f1 * Z2.z + f2 * AZ.z - f3 * Z1.z;
    Z3.w = f1 * Z2.w + f2 * AZ.w - f3 * Z1.w;

    float4 o;
    o.x = 0.25f * (coef0 * X.x + coef1 * Z1.x + coef2 * Z2.x + coef3 * Z3.x);
    o.y = 0.25f * (coef0 * X.y + coef1 * Z1.y + coef2 * Z2.y + coef3 * Z3.y);
    o.z = 0.25f * (coef0 * X.z + coef1 * Z1.z + coef2 * Z2.z + coef3 * Z3.z);
    o.w = 0.25f * (coef0 * X.w + coef1 * Z1.w + coef2 * Z2.w + coef3 * Z3.w);
    out[i] = o;
}

// ---------------------------------------------------------------------------
extern "C" void kernel_launch(void* const* d_in, const int* in_sizes, int n_in,
                              void* d_out, int out_size, void* d_ws, size_t ws_size,
                              hipStream_t stream)
{
    const float* x   = (const float*)d_in[0];
    const int*   ei  = (const int*)d_in[1];   // [2, E] flat: rows then cols
    const float* w   = (const float*)d_in[2];
    const float* gam = (const float*)d_in[3]; // [L+1, 1] = 4 floats
    float*       out = (float*)d_out;

    const int E  = in_sizes[1] / 2;
    const int ND = in_sizes[0];               // N * 64
    const int n4 = ND / 4;

    const int* row = ei;
    const int* col = ei + E;

    float* z1 = (float*)d_ws;                 // workspace: 3 * ND floats
    float* z2 = z1 + ND;
    float* Az = z2 + ND;

    const dim3 blkE(BLOCK), grdE((E + TILE_E - 1) / TILE_E);
    const dim3 blkV(256),   grdV((n4 + 255) / 256);
    const size_t bytesND = (size_t)ND * sizeof(float);

    // ---- l = 1: z1 = ((a-b)/2) x + ((a+b+2)/2) A x ----
    hipMemsetAsync(Az, 0, bytesND, stream);
    spmm_kernel<<<grdE, blkE, 0, stream>>>(row, col, w, x, Az, E);
    combine3_kernel<<<grdV, blkV, 0, stream>>>(
        (const float4*)Az, (const float4*)x, (const float4*)x, (float4*)z1,
        (float)((JA + JB + 2.0) / 2.0), (float)((JA - JB) / 2.0), 0.0f, n4);

    // ---- l = 2: z2 = (jc1*z1 + jc2*A z1 - jc3*x) / jc0 ----
    hipMemsetAsync(Az, 0, bytesND, stream);
    spmm_kernel<<<grdE, blkE, 0, stream>>>(row, col, w, z1, Az, E);
    combine3_kernel<<<grdV, blkV, 0, stream>>>(
        (const float4*)Az, (const float4*)x, (const float4*)z1, (float4*)z2,
        (float)(jc2(2) / jc0(2)), (float)(-jc3(2) / jc0(2)), (float)(jc1(2) / jc0(2)), n4);

    // ---- l = 3 fused into final combine ----
    hipMemsetAsync(Az, 0, bytesND, stream);
    spmm_kernel<<<grdE, blkE, 0, stream>>>(row, col, w, z2, Az, E);
    final_kernel<<<grdV, blkV, 0, stream>>>(
        (const float4*)x, (const float4*)z1, (const float4*)z2, (const float4*)Az,
        gam, (float4*)out, n4);
}